// HierarchicalLogLikelihoodLossLayer_64707977281687
// MI455X (gfx1250) — compile-verified
//
#include <hip/hip_runtime.h>
#include <math.h>

#define NSTATES 12
#define NCH     64
#define BATCH   32
#define SEQ     1024
#define NSUBJ   100

typedef __attribute__((ext_vector_type(2))) float v2f;
typedef __attribute__((ext_vector_type(8))) float v8f;

// ---------------- zero ws accumulators (re-run every call) ----------------
__global__ void k_zero(float* regp) { regp[0] = 0.f; regp[1] = 0.f; }

// ---------------- per-subject regularizer ----------------
__global__ void __launch_bounds__(256)
k_reg(const float* mu_pop, const float* L_pop,
      const float* mu_subj, const float* L_subj,
      const int* sid, float* regp) {
  __shared__ int pres;
  __shared__ float red[256];
  const int s = blockIdx.x, tid = threadIdx.x;
  if (tid == 0) pres = 0;
  __syncthreads();
  if (tid < BATCH && sid[tid] == s) atomicOr(&pres, 1);
  __syncthreads();
  if (!pres) return;                       // uniform exit
  float acc = 0.f;
  const int MUE = NSTATES * NCH;
  for (int e = tid; e < MUE; e += 256) {
    float d = mu_subj[(size_t)s * MUE + e] - mu_pop[e];
    acc += d * d;
  }
  const int LE = NSTATES * NCH * NCH;
  for (int e = tid; e < LE; e += 256) {
    float d = L_subj[(size_t)s * LE + e] - L_pop[e];
    acc += d * d;
  }
  red[tid] = acc; __syncthreads();
  for (int off = 128; off > 0; off >>= 1) {
    if (tid < off) red[tid] += red[tid + off];
    __syncthreads();
  }
  if (tid == 0) {
    atomicAdd(&regp[0], 0.5f * red[0]);    // LAMBDA_MU/2 == LAMBDA_L/2 == 0.5
    atomicAdd(&regp[1], 1.0f);             // present count
  }
}

__global__ void k_reg_final(const float* regp, float* out) {
  out[0] = regp[0] * regp[1] * (1.0f / (float)NSUBJ);   // fresh init of d_out
}

// ---------------- per-(b,k): Linv, w = Linv*mu, logdet ----------------
__global__ void __launch_bounds__(64)
k_prep(const float* mu_subj, const float* L_subj, const int* sid,
       float* Linv_g, float* w_g, float* ld_g) {
  const int bk = blockIdx.x;
  const int b = bk / NSTATES, k = bk % NSTATES;
  const int s = sid[b];
  const float* Lsrc = L_subj + (size_t)(s * NSTATES + k) * NCH * NCH;
  const float* msrc = mu_subj + (size_t)(s * NSTATES + k) * NCH;

  __shared__ float Ls[NCH][NCH + 1];
  __shared__ float Li[NCH][NCH + 1];
  __shared__ float muS[NCH];
  __shared__ float red[NCH];
  const int tid = threadIdx.x;             // 64 threads

  for (int c = 0; c < NCH; c += 4) {
    float4 v = *(const float4*)(Lsrc + (size_t)tid * NCH + c);
    Ls[tid][c] = v.x; Ls[tid][c + 1] = v.y; Ls[tid][c + 2] = v.z; Ls[tid][c + 3] = v.w;
  }
  muS[tid] = msrc[tid];
  __syncthreads();

  // thread j owns column j of Linv (forward substitution, no cross-thread deps)
  const int j = tid;
  for (int i = 0; i < j; i++) Li[i][j] = 0.f;
  Li[j][j] = 1.0f / Ls[j][j];
  for (int i = j + 1; i < NCH; i++) {
    float sum = 0.f;
    for (int m = j; m < i; m++) sum += Ls[i][m] * Li[m][j];
    Li[i][j] = -sum / Ls[i][i];
  }
  red[j] = logf(Ls[j][j]);
  __syncthreads();

  float* dst = Linv_g + (size_t)bk * NCH * NCH + (size_t)tid * NCH;
  for (int c = 0; c < NCH; c += 4) {
    float4 v = make_float4(Li[tid][c], Li[tid][c + 1], Li[tid][c + 2], Li[tid][c + 3]);
    *(float4*)(dst + c) = v;
  }
  float wv = 0.f;
  for (int m = 0; m <= tid; m++) wv += Li[tid][m] * muS[m];
  w_g[(size_t)bk * NCH + tid] = wv;
  if (tid == 0) {
    float ldet = 0.f;
    for (int j2 = 0; j2 < NCH; j2++) ldet += red[j2];
    ld_g[bk] = ldet;
  }
}

// ---------------- main: z = Linv * X via WMMA fp32, fused (z-w)^2, gamma ----------------
// LDS layout: K-pair interleaved + XOR-16 parity swizzle so every WMMA fragment
// is ONE aligned ds_load_b64 into an even VGPR pair (no repacking movs).
//   element (c, col) lives at  arr[c>>1][ col ^ ((c>>1 & 1) << 4) ]  as v2f {c, c+1}
__global__ void __launch_bounds__(256)
k_main(const float* data, const float* gamma, const float* Linv_g,
       const float* w_g, const float* ld_g, float* out) {
  __shared__ v2f Xs2[NCH / 2][NCH];        // data tile (K-pair, t_local)
  __shared__ v2f Ls2[NCH / 2][NCH];        // Linv     (K-pair, m row)
  __shared__ float quadS[NCH];
  __shared__ float wlds[NCH];

  const int b  = blockIdx.y;
  const int t0 = blockIdx.x * 64;
  const int tid  = threadIdx.x;
  const int lane = tid & 31, wave = tid >> 5;
  const int hi = lane >> 4, l16 = lane & 15;
  const int sx = hi << 4;                  // fragment-side swizzle (pair parity == hi)

  {  // stage data tile: pair-interleave + swizzle; b64 LDS stores
    const int tl = tid & 63, cg = tid >> 6;
    const float* src = data + ((size_t)b * SEQ + t0 + tl) * NCH + cg * 16;
    for (int i = 0; i < 4; i++) {
      float4 v = *(const float4*)(src + i * 4);
      const int p0 = (cg * 16 + i * 4) >> 1;     // even pair
      v2f lo = {v.x, v.y}, hi2 = {v.z, v.w};
      Xs2[p0][tl]      = lo;                     // swz(even, tl) == tl
      Xs2[p0 + 1][tl ^ 16] = hi2;                // odd pair -> XOR 16
    }
  }

  // wave's two output tiles share one m-tile; two t-tiles -> shared A-fragment
  const int m0  = (wave >> 1) * 16;
  const int ta  = ((wave & 1) * 2 + 0) * 16;
  const int tb  = ((wave & 1) * 2 + 1) * 16;

  float wsum = 0.f;                               // live for tid < 64
  const float HALF_C_LOG2PI = 58.8120662f;        // 0.5 * 64 * ln(2*pi)

  for (int k = 0; k < NSTATES; k++) {
    const int bk = b * NSTATES + k;
    __syncthreads();                              // quadS/Ls2/wlds reuse fence
    if (tid < NCH) { quadS[tid] = 0.f; wlds[tid] = w_g[(size_t)bk * NCH + tid]; }
    {
      const float4* src = (const float4*)(Linv_g + (size_t)bk * NCH * NCH);
      for (int i = 0; i < 4; i++) {
        float4 v = src[tid * 4 + i];
        const int e = tid * 16 + i * 4;
        const int m = e >> 6, p0 = (e & 63) >> 1; // p0 even
        v2f lo = {v.x, v.y}, hi2 = {v.z, v.w};
        Ls2[p0][m]      = lo;
        Ls2[p0 + 1][m ^ 16] = hi2;
      }
    }
    if (k + 1 < NSTATES)   // pull next state's Linv toward L2/L0 (global_prefetch_b8)
      __builtin_prefetch(Linv_g + (size_t)(bk + 1) * NCH * NCH + tid * 16, 0, 1);
    __syncthreads();

    // dual-accumulator kstep loop: one A-fragment feeds two independent WMMA chains
    v8f acc0 = {};
    v8f acc1 = {};
    for (int ks = 0; ks < 16; ks++) {
      const int cp = ks * 2 + hi;                 // K-pair index; cp&1 == hi
      // A (16x4 f32): lanes 0-15 hold K0/K1, lanes 16-31 hold K2/K3
      v2f a  = Ls2[cp][(m0 + l16) ^ sx];          // one ds_load_b64
      // B (4x16 f32): mirrored K split across lane halves
      v2f b0 = Xs2[cp][(ta + l16) ^ sx];          // one ds_load_b64
      v2f b1 = Xs2[cp][(tb + l16) ^ sx];          // one ds_load_b64
      acc0 = __builtin_amdgcn_wmma_f32_16x16x4_f32(
          false, a, false, b0, (short)0, acc0, false, false);
      acc1 = __builtin_amdgcn_wmma_f32_16x16x4_f32(
          false, a, false, b1, (short)0, acc1, false, false);
    }
    // lane holds rows m0 + r + 8*hi; columns ta+l16 (acc0) and tb+l16 (acc1)
    float part0 = 0.f, part1 = 0.f;
    for (int r = 0; r < 8; r++) {
      const float wr = wlds[m0 + r + 8 * hi];
      float v0 = acc0[r] - wr;
      float v1 = acc1[r] - wr;
      part0 += v0 * v0;
      part1 += v1 * v1;
    }
    atomicAdd(&quadS[ta + l16], part0);           // ds_add_f32
    atomicAdd(&quadS[tb + l16], part1);
    __syncthreads();

    if (tid < NCH) {
      const float g  = gamma[((size_t)b * SEQ + t0 + tid) * NSTATES + k];
      const float lp = -0.5f * quadS[tid] - ld_g[bk] - HALF_C_LOG2PI;
      wsum += g * lp;
    }
  }
  if (tid < NCH)
    atomicAdd(out, -wsum * (1.0f / (float)BATCH)); // nll on top of reg
}

extern "C" void kernel_launch(void* const* d_in, const int* in_sizes, int n_in,
                              void* d_out, int out_size, void* d_ws, size_t ws_size,
                              hipStream_t stream) {
  const float* data    = (const float*)d_in[0];
  const float* mu_pop  = (const float*)d_in[1];
  const float* L_pop   = (const float*)d_in[2];
  const float* mu_subj = (const float*)d_in[3];
  const float* L_subj  = (const float*)d_in[4];
  const float* gamma   = (const float*)d_in[5];
  const int*   sid     = (const int*)d_in[6];
  float* out = (float*)d_out;

  // ws layout (floats): Linv [384*4096] | w [384*64] | logdet [384] | regp [2]  (~6.4 MB)
  float* Linv_g = (float*)d_ws;
  float* w_g    = Linv_g + (size_t)BATCH * NSTATES * NCH * NCH;
  float* ld_g   = w_g + (size_t)BATCH * NSTATES * NCH;
  float* regp   = ld_g + BATCH * NSTATES;

  k_zero<<<1, 1, 0, stream>>>(regp);
  k_reg<<<NSUBJ, 256, 0, stream>>>(mu_pop, L_pop, mu_subj, L_subj, sid, regp);
  k_reg_final<<<1, 1, 0, stream>>>(regp, out);
  k_prep<<<BATCH * NSTATES, 64, 0, stream>>>(mu_subj, L_subj, sid, Linv_g, w_g, ld_g);
  dim3 grid(SEQ / 64, BATCH);
  k_main<<<grid, 256, 0, stream>>>(data, gamma, Linv_g, w_g, ld_g, out);
}